// Gcn_Attention_28398323761501
// MI455X (gfx1250) — compile-verified
//
#include <hip/hip_runtime.h>
#include <hip/hip_bf16.h>

// ---------------------------------------------------------------------------
// GCN block for MI455X (gfx1250, wave32).
//   h   = relu(x @ W1 + b1)                       [N,64]
//   deg = 1 + scatter-count(dst);  dinv = rsqrt(deg)
//   hg  = h @ Wg              <-- V_WMMA_F32_16X16X4_F32 (fp32 tensor path)
//   agg = scatter_add(hg[src] * dinv[src]*dinv[dst], dst) (+ self loop)
//   out = LayerNorm_128(concat(h, relu(agg + bg))) * gamma + beta
// ---------------------------------------------------------------------------

typedef __attribute__((ext_vector_type(2))) float v2f;
typedef __attribute__((ext_vector_type(8))) float v8f;

#define DOUT 64
#define EPS 1e-5f

// --- kernel 1: init deg=1 (self loop) and agg=0 ----------------------------
__global__ void init_kernel(float* __restrict__ deg, float* __restrict__ agg, int n) {
    int i = blockIdx.x * blockDim.x + threadIdx.x;
    if (i < n * DOUT) agg[i] = 0.0f;
    if (i < n)        deg[i] = 1.0f;
}

// --- kernel 2: h = relu(x @ W1 + b1), DIN=3 --------------------------------
__global__ void lin1_kernel(const float* __restrict__ x, const float* __restrict__ W1,
                            const float* __restrict__ b1, float* __restrict__ h, int n) {
    int i = blockIdx.x * blockDim.x + threadIdx.x;      // node*64 + col
    if (i >= n * DOUT) return;
    int node = i >> 6, c = i & 63;
    float v = x[node * 3 + 0] * W1[0 * DOUT + c]
            + x[node * 3 + 1] * W1[1 * DOUT + c]
            + x[node * 3 + 2] * W1[2 * DOUT + c] + b1[c];
    h[i] = v > 0.0f ? v : 0.0f;
}

// --- kernel 3: degree accumulation over dst --------------------------------
__global__ void deg_kernel(const long long* __restrict__ ei, float* __restrict__ deg,
                           int n_edges) {
    int e = blockIdx.x * blockDim.x + threadIdx.x;
    if (e >= n_edges) return;
    int dst = (int)ei[n_edges + e];
    atomicAdd(&deg[dst], 1.0f);
}

// --- kernel 4: dinv = rsqrt(deg)  (deg >= 1 always, self loop) -------------
__global__ void dinv_kernel(const float* __restrict__ deg, float* __restrict__ dinv, int n) {
    int i = blockIdx.x * blockDim.x + threadIdx.x;
    if (i < n) dinv[i] = __frsqrt_rn(deg[i]);
}

// --- kernel 5: hg = h @ Wg via V_WMMA_F32_16X16X4_F32 ----------------------
// Block = 256 threads = 8 waves. Block computes 32 rows x 64 cols.
// Wave w: row-tile rt = w>>2 (16 rows), col-tile ct = w&3 (16 cols).
__global__ void gemm_wmma_kernel(const float* __restrict__ h, const float* __restrict__ Wg,
                                 float* __restrict__ hg, int n) {
    __shared__ float sB[DOUT * DOUT];   // 64x64 Wg, 16 KB
    __shared__ float sA[32 * DOUT];     // 32-row h tile, 8 KB
    const int tid  = threadIdx.x;
    const int wave = tid >> 5;
    const int lane = tid & 31;
    const int rowbase = blockIdx.x * 32;

    // stage Wg (4096 floats) and the 32x64 h tile (2048 floats) into LDS
    for (int i = tid; i < DOUT * DOUT; i += 256) sB[i] = Wg[i];
    for (int i = tid; i < 32 * DOUT; i += 256) {
        int r = rowbase + (i >> 6);
        sA[i] = (r < n) ? h[r * DOUT + (i & 63)] : 0.0f;
    }
    __syncthreads();

    const int rt = wave >> 2;           // 0..1
    const int ct = wave & 3;            // 0..3
    const int m  = lane & 15;           // row (A) / col (B) within tile
    const int kk = (lane >> 4) * 2;     // K sub-stripe: lanes 0-15 -> {0,1}, 16-31 -> {2,3}

    v8f acc = {};
#pragma unroll
    for (int k = 0; k < DOUT; k += 4) {
        v2f a, b;
        a.x = sA[(rt * 16 + m) * DOUT + k + kk];
        a.y = sA[(rt * 16 + m) * DOUT + k + kk + 1];
        b.x = sB[(k + kk) * DOUT + ct * 16 + m];
        b.y = sB[(k + kk + 1) * DOUT + ct * 16 + m];
        // D = A(16x4) x B(4x16) + C ; fp32 accumulate, EXEC all-ones here
        acc = __builtin_amdgcn_wmma_f32_16x16x4_f32(
            /*neg_a=*/false, a, /*neg_b=*/false, b,
            /*c_mod=*/(short)0, acc, /*reuse_a=*/false, /*reuse_b=*/false);
    }

    // C/D layout: VGPR v -> M = v + (lane>=16 ? 8 : 0), N = lane&15
    const int row0 = rowbase + rt * 16 + (lane >> 4) * 8;
    const int col  = ct * 16 + (lane & 15);
#pragma unroll
    for (int v = 0; v < 8; ++v) {
        int r = row0 + v;
        if (r < n) hg[r * DOUT + col] = acc[v];
    }
}

// --- kernel 6: edge scatter: agg[dst] += hg[src] * dinv[src]*dinv[dst] -----
// 16 threads per edge, float4 per thread (64 cols).
__global__ void scatter_kernel(const long long* __restrict__ ei, const float* __restrict__ hg,
                               const float* __restrict__ dinv, float* __restrict__ agg,
                               int n_edges) {
    long long idx = (long long)blockIdx.x * blockDim.x + threadIdx.x;
    if (idx >= (long long)n_edges * 16) return;
    int e = (int)(idx >> 4);
    int g = ((int)idx & 15) * 4;
    int src = (int)ei[e];
    int dst = (int)ei[n_edges + e];
    float nrm = dinv[src] * dinv[dst];
    const float4 v = *(const float4*)(hg + (long long)src * DOUT + g);
    float* o = agg + (long long)dst * DOUT + g;
    atomicAdd(o + 0, v.x * nrm);
    atomicAdd(o + 1, v.y * nrm);
    atomicAdd(o + 2, v.z * nrm);
    atomicAdd(o + 3, v.w * nrm);
}

// --- kernel 7: self loop + bias + relu + concat + LayerNorm(128) -----------
// One wave32 per node; lane l owns 4 of the 128 concat values.
__global__ void final_kernel(const float* __restrict__ h, const float* __restrict__ hg,
                             const float* __restrict__ agg, const float* __restrict__ dinv,
                             const float* __restrict__ bg, const float* __restrict__ gamma,
                             const float* __restrict__ beta, float* __restrict__ out, int n) {
    int node = blockIdx.x * (blockDim.x / 32) + (threadIdx.x >> 5);
    int lane = threadIdx.x & 31;
    if (node >= n) return;

    float d2 = dinv[node] * dinv[node];   // self-loop norm
    float vals[4];
#pragma unroll
    for (int j = 0; j < 4; ++j) {
        int c = lane * 4 + j;             // concat column 0..127
        float v;
        if (c < DOUT) {
            v = h[(long long)node * DOUT + c];
        } else {
            int cc = c - DOUT;
            v = agg[(long long)node * DOUT + cc]
              + hg[(long long)node * DOUT + cc] * d2 + bg[cc];
            v = v > 0.0f ? v : 0.0f;      // relu
        }
        vals[j] = v;
    }
    float s = vals[0] + vals[1] + vals[2] + vals[3];
    float q = vals[0]*vals[0] + vals[1]*vals[1] + vals[2]*vals[2] + vals[3]*vals[3];
#pragma unroll
    for (int off = 16; off > 0; off >>= 1) {   // wave32 butterfly
        s += __shfl_xor(s, off, 32);
        q += __shfl_xor(q, off, 32);
    }
    float mu  = s * (1.0f / 128.0f);
    float var = q * (1.0f / 128.0f) - mu * mu;
    float rs  = __frsqrt_rn(var + EPS);
#pragma unroll
    for (int j = 0; j < 4; ++j) {
        int c = lane * 4 + j;
        out[(long long)node * 128 + c] = (vals[j] - mu) * rs * gamma[c] + beta[c];
    }
}

// ---------------------------------------------------------------------------
extern "C" void kernel_launch(void* const* d_in, const int* in_sizes, int n_in,
                              void* d_out, int out_size, void* d_ws, size_t ws_size,
                              hipStream_t stream) {
    const float*     x     = (const float*)d_in[0];
    const long long* ei    = (const long long*)d_in[1];   // int64 edge_index [2,E]
    const float*     W1    = (const float*)d_in[2];
    const float*     b1    = (const float*)d_in[3];
    const float*     Wg    = (const float*)d_in[4];
    const float*     bg    = (const float*)d_in[5];
    const float*     gamma = (const float*)d_in[6];
    const float*     beta  = (const float*)d_in[7];
    float*           out   = (float*)d_out;

    const int n = in_sizes[0] / 3;       // nodes
    const int e = in_sizes[1] / 2;       // edges

    // workspace carve (all fp32, 256B-aligned chunks)
    char* ws = (char*)d_ws;
    size_t off = 0;
    auto carve = [&](size_t bytes) { char* p = ws + off; off = (off + bytes + 255) & ~(size_t)255; return p; };
    float* h    = (float*)carve((size_t)n * DOUT * sizeof(float));
    float* hg   = (float*)carve((size_t)n * DOUT * sizeof(float));
    float* agg  = (float*)carve((size_t)n * DOUT * sizeof(float));
    float* deg  = (float*)carve((size_t)n * sizeof(float));
    float* dinv = (float*)carve((size_t)n * sizeof(float));

    const int B = 256;
    init_kernel<<<(n * DOUT + B - 1) / B, B, 0, stream>>>(deg, agg, n);
    lin1_kernel<<<(n * DOUT + B - 1) / B, B, 0, stream>>>(x, W1, b1, h, n);
    deg_kernel<<<(e + B - 1) / B, B, 0, stream>>>(ei, deg, e);
    dinv_kernel<<<(n + B - 1) / B, B, 0, stream>>>(deg, dinv, n);
    gemm_wmma_kernel<<<(n + 31) / 32, B, 0, stream>>>(h, Wg, hg, n);
    long long sthreads = (long long)e * 16;
    scatter_kernel<<<(unsigned)((sthreads + B - 1) / B), B, 0, stream>>>(ei, hg, dinv, agg, e);
    final_kernel<<<(n + 7) / 8, B, 0, stream>>>(h, hg, agg, dinv, bg, gamma, beta, out, n);
}